// relation_net_multi_56650618634365
// MI455X (gfx1250) — compile-verified
//
#include <hip/hip_runtime.h>
#include <math.h>

typedef __attribute__((ext_vector_type(2))) float v2f;
typedef __attribute__((ext_vector_type(8))) float v8f;

#define B_   4
#define N_   8192
#define DIN  32
#define DOUT 64
#define EPSV 1e-5f

// ---------------------------------------------------------------------------
// K1: fused triple GEMM with V_WMMA_F32_16X16X4_F32.
//   Y  [b][e][n]  = fs_w  (64x32) @ vf[b] (32xN) + fs_b   (pre-BN)
//   Z0 [b][n][e]  = fr_w0 (64x32) @ vf[b]                  (gather-friendly layout)
//   Z1 [b][n][e]  = fr_w1 (64x32) @ vf[b]
// One wave per 16(e) x 16(n) tile; K=32 done as 8 WMMA steps of K=4.
// ---------------------------------------------------------------------------
__global__ __launch_bounds__(256) void k_gemm3(
    const float* __restrict__ vf, const float* __restrict__ fs_w,
    const float* __restrict__ fr_w0, const float* __restrict__ fr_w1,
    const float* __restrict__ fs_b,
    float* __restrict__ Y, float* __restrict__ Z0, float* __restrict__ Z1)
{
    const int lane = threadIdx.x & 31;
    const int wave = threadIdx.x >> 5;
    const int tile = blockIdx.x * 8 + wave;          // 24576 tiles total
    const int nt  = tile & 511;                      // 512 n-tiles
    const int et  = (tile >> 9) & 3;                 // 4 e-tiles
    const int mat = (tile >> 11) % 3;                // which output matrix
    const int b   = tile / 6144;                     // batch
    const float* W = (mat == 0) ? fs_w : ((mat == 1) ? fr_w0 : fr_w1);

    const int col = lane & 15;                       // also A's M row within tile
    const int hi  = lane >> 4;
    const int n0  = nt << 4;
    const int e0  = et << 4;
    const float* vfb = vf + (size_t)b * DIN * N_;

    v8f c = {0.f, 0.f, 0.f, 0.f, 0.f, 0.f, 0.f, 0.f};
#pragma unroll
    for (int k = 0; k < 8; ++k) {
        const int d0 = 4 * k + 2 * hi;
        v2f a, bb;
        a.x  = W[(e0 + col) * DIN + d0];
        a.y  = W[(e0 + col) * DIN + d0 + 1];
        bb.x = vfb[(size_t)d0 * N_ + n0 + col];
        bb.y = vfb[(size_t)(d0 + 1) * N_ + n0 + col];
        c = __builtin_amdgcn_wmma_f32_16x16x4_f32(
                false, a, false, bb, (short)0, c, false, false);
    }

    const int n = n0 + col;
    if (mat == 0) {
#pragma unroll
        for (int v = 0; v < 8; ++v) {
            const int e = e0 + v + 8 * hi;
            Y[((size_t)b * DOUT + e) * N_ + n] = c[v] + fs_b[e];
        }
    } else {
        float* Z = (mat == 1) ? Z0 : Z1;
        float* dst = Z + ((size_t)(b * N_ + n)) * DOUT + e0 + 8 * hi;
#pragma unroll
        for (int v = 0; v < 8; ++v) dst[v] = c[v];   // 32B contiguous per lane
    }
}

// ---------------------------------------------------------------------------
// K2: per-channel sum / sumsq of Y over (b, n). One block per channel.
// ---------------------------------------------------------------------------
__global__ __launch_bounds__(256) void k_fs_stats(const float* __restrict__ Y,
                                                  float* __restrict__ stats)
{
    __shared__ float ssum[256], ssq[256];
    const int e = blockIdx.x, tid = threadIdx.x;
    float s = 0.f, q = 0.f;
    for (int b = 0; b < B_; ++b) {
        const float* p = Y + ((size_t)b * DOUT + e) * N_;
        for (int i = tid; i < N_; i += 256) { float v = p[i]; s += v; q += v * v; }
    }
    ssum[tid] = s; ssq[tid] = q; __syncthreads();
    for (int off = 128; off > 0; off >>= 1) {
        if (tid < off) { ssum[tid] += ssum[tid + off]; ssq[tid] += ssq[tid + off]; }
        __syncthreads();
    }
    if (tid == 0) { stats[e] = ssum[0]; stats[DOUT + e] = ssq[0]; }
}

// ---------------------------------------------------------------------------
// K3: base features = relu(BN(Y)) -> d_out (B,64,N)
// ---------------------------------------------------------------------------
__global__ __launch_bounds__(256) void k_base(const float* __restrict__ Y,
                                              const float* __restrict__ stats,
                                              const float* __restrict__ g,
                                              const float* __restrict__ be,
                                              float* __restrict__ out)
{
    const size_t i = (size_t)blockIdx.x * 256 + threadIdx.x;
    const int e = (int)((i / N_) & (DOUT - 1));
    const float cnt = (float)(B_ * N_);
    const float m   = stats[e] / cnt;
    const float var = stats[DOUT + e] / cnt - m * m;
    const float inv = rsqrtf(var + EPSV);
    const float r = (Y[i] - m) * inv * g[e] + be[e];
    out[i] = fmaxf(r, 0.f);
}

// ---------------------------------------------------------------------------
// K4: exact KNN with ASYNC double-buffered candidate tiles.
// Candidate coords stream into LDS via GLOBAL_LOAD_ASYNC_TO_LDS_B128
// (ASYNCcnt-tracked; no VGPR round trip): while tile t is scanned, tile t+1
// is in flight into the other buffer; s_wait_asynccnt 6 releases tile t
// without draining the prefetch (6 = own wave's in-flight ops for tile t+1).
// One thread per query; per-thread sorted top-32 lists in LDS.
// Slot 16 overwritten with self index (reference's .at[:,:,0].set(self)).
// ---------------------------------------------------------------------------
#define QT 128
#define CT 1024
__global__ __launch_bounds__(QT) void k_knn(const float* __restrict__ vc,
                                            int* __restrict__ knn)
{
    __shared__ __align__(16) float sbuf[2][3 * CT];  // [buf][x|y|z blocks]
    __shared__ float sd[32 * QT];
    __shared__ int   si[32 * QT];
    const int tid = threadIdx.x;
    const int bpb = N_ / QT;
    const int b = blockIdx.x / bpb;
    const int n = (blockIdx.x % bpb) * QT + tid;

    const float* cx = vc + (size_t)b * 3 * N_;
    const float qx = cx[n], qy = cx[N_ + n], qz = cx[2 * N_ + n];
    const float qq = qx * qx + qy * qy + qz * qz;

    // 32-bit LDS byte address of sbuf (low 32 bits of generic shared pointer)
    const unsigned sbase = (unsigned)(size_t)(&sbuf[0][0]);

#pragma unroll
    for (int s = 0; s < 32; ++s) { sd[s * QT + tid] = 3.4e38f; si[s * QT + tid] = 0; }
    float worst = 3.4e38f;

    // issue tile 0 into buffer 0 ---------------------------------------------
    {
        const int base = 0;
        const unsigned q16 = (unsigned)tid * 16u;
#pragma unroll
        for (int c = 0; c < 3; ++c) {
            const float* gp = cx + (size_t)c * N_ + base;
#pragma unroll
            for (int r = 0; r < 2; ++r) {
                const unsigned voff  = q16 + (unsigned)r * (QT * 16);
                const unsigned laddr = sbase + (unsigned)c * (CT * 4) + voff;
                asm volatile("global_load_async_to_lds_b128 %0, %1, %2"
                             :: "v"(laddr), "v"(voff), "s"(gp) : "memory");
            }
        }
    }

    const int ntiles = N_ / CT;                      // 8
    for (int t = 0; t < ntiles; ++t) {
        const int buf = t & 1;
        if (t + 1 < ntiles) {                        // prefetch next tile
            const int base = (t + 1) * CT;
            const unsigned q16 = (unsigned)tid * 16u;
            const unsigned bofs = sbase + (unsigned)(buf ^ 1) * (3 * CT * 4);
#pragma unroll
            for (int c = 0; c < 3; ++c) {
                const float* gp = cx + (size_t)c * N_ + base;
#pragma unroll
                for (int r = 0; r < 2; ++r) {
                    const unsigned voff  = q16 + (unsigned)r * (QT * 16);
                    const unsigned laddr = bofs + (unsigned)c * (CT * 4) + voff;
                    asm volatile("global_load_async_to_lds_b128 %0, %1, %2"
                                 :: "v"(laddr), "v"(voff), "s"(gp) : "memory");
                }
            }
            asm volatile("s_wait_asynccnt 0x6" ::: "memory");
        } else {
            asm volatile("s_wait_asynccnt 0x0" ::: "memory");
        }
        __syncthreads();                             // tile t visible to block

        const float* sx = &sbuf[buf][0];
        const float* sy = sx + CT;
        const float* sz = sy + CT;
        const int base = t * CT;
        for (int j = 0; j < CT; ++j) {
            const float x = sx[j], y = sy[j], z = sz[j];
            const float cc  = x * x + y * y + z * z;
            const float dot = qx * x + qy * y + qz * z;
            const float d = qq + cc - 2.f * dot;
            if (d < worst) {
                int p = 31;
                while (p > 0 && sd[(p - 1) * QT + tid] > d) {
                    sd[p * QT + tid] = sd[(p - 1) * QT + tid];
                    si[p * QT + tid] = si[(p - 1) * QT + tid];
                    --p;
                }
                sd[p * QT + tid] = d;
                si[p * QT + tid] = base + j;
                worst = sd[31 * QT + tid];
            }
        }
        __syncthreads();                             // done with this buffer
    }

    int* dst = knn + ((size_t)(b * N_ + n)) * 32;
#pragma unroll
    for (int s = 0; s < 32; ++s) dst[s] = si[s * QT + tid];
    dst[16] = si[tid];                               // self index for layer 1
}

// ---------------------------------------------------------------------------
// K5a: layer statistics. One wave per point (8 points/block).
//  - h: sum/sumsq per channel e of gathered Z rows       (hstats[0..127])
//  - t: sum/sumsq per o of t[o,k] = cw[o]·[sc, sc-cn_k]  (tstats[0..31])
// Block-local accumulation via LDS ds_add_f32, then global atomics.
// ---------------------------------------------------------------------------
__global__ __launch_bounds__(256) void k_layer_stats(
    const float* __restrict__ Z, const int* __restrict__ knn,
    const float* __restrict__ vc, const float* __restrict__ cw, int kbase,
    float* __restrict__ hstats, float* __restrict__ tstats)
{
    __shared__ float hsum[DOUT], hsq[DOUT], tsum[16], tsq[16];
    const int tid = threadIdx.x, lane = tid & 31, wave = tid >> 5;
    if (tid < DOUT) { hsum[tid] = 0.f; hsq[tid] = 0.f; }
    if (tid < 16)   { tsum[tid] = 0.f; tsq[tid]  = 0.f; }
    __syncthreads();

    const int p = blockIdx.x * 8 + wave;
    const int b = p / N_;
    const int jreg = knn[(size_t)p * 32 + kbase + (lane & 15)];

    // --- h part: lane covers channels lane and lane+32 ---
    const float* Zb = Z + (size_t)b * N_ * DOUT;
    float s0 = 0.f, q0 = 0.f, s1 = 0.f, q1 = 0.f;
#pragma unroll
    for (int k = 0; k < 16; ++k) {
        const int j = __shfl(jreg, k, 32);
        const float* zr = Zb + (size_t)j * DOUT;
        const float v0 = zr[lane], v1 = zr[lane + 32];
        s0 += v0; q0 += v0 * v0; s1 += v1; q1 += v1 * v1;
    }
    atomicAdd(&hsum[lane], s0);       atomicAdd(&hsq[lane], q0);
    atomicAdd(&hsum[lane + 32], s1);  atomicAdd(&hsq[lane + 32], q1);

    // --- t part: lane -> o = lane&15, k half = (lane>>4)*8 .. +7 ---
    const float* cx = vc + (size_t)b * 3 * N_;
    const int j0 = __shfl(jreg, 0, 32);
    const float scx = cx[j0], scy = cx[N_ + j0], scz = cx[2 * N_ + j0];
    const int o = lane & 15;
    const float* w = cw + o * 6;
    const float w3 = w[3], w4 = w[4], w5 = w[5];
    const float tb = w[0] * scx + w[1] * scy + w[2] * scz
                   + w3 * scx + w4 * scy + w5 * scz;
    float ts = 0.f, tq = 0.f;
    const int k0 = (lane >> 4) * 8;
#pragma unroll
    for (int kk = 0; kk < 8; ++kk) {
        const int j = __shfl(jreg, k0 + kk, 32);
        const float t = tb - (w3 * cx[j] + w4 * cx[N_ + j] + w5 * cx[2 * N_ + j]);
        ts += t; tq += t * t;
    }
    ts += __shfl_xor(ts, 16, 32);
    tq += __shfl_xor(tq, 16, 32);
    if (lane < 16) { atomicAdd(&tsum[o], ts); atomicAdd(&tsq[o], tq); }
    __syncthreads();
    if (tid < DOUT) { atomicAdd(&hstats[tid], hsum[tid]); atomicAdd(&hstats[DOUT + tid], hsq[tid]); }
    if (tid < 16)   { atomicAdd(&tstats[tid], tsum[tid]); atomicAdd(&tstats[16 + tid],  tsq[tid]); }
}

// ---------------------------------------------------------------------------
// K5b: apply layer. One wave per point.
//   w[k] = sum_o tanh(BN(t[o,k]))   (lanes split o-range, combined via shfl)
//   out[b,e,n] += sum_k w[k] * relu(BN(Zg[k,e]))
// ---------------------------------------------------------------------------
__global__ __launch_bounds__(256) void k_layer_apply(
    const float* __restrict__ Z, const int* __restrict__ knn,
    const float* __restrict__ vc, const float* __restrict__ cw,
    const float* __restrict__ cg, const float* __restrict__ cb,
    const float* __restrict__ fg, const float* __restrict__ fb,
    const float* __restrict__ hstats, const float* __restrict__ tstats,
    int kbase, float* __restrict__ out)
{
    const int tid = threadIdx.x, lane = tid & 31, wave = tid >> 5;
    const int p = blockIdx.x * 8 + wave;
    const int b = p / N_, n = p % N_;
    const int jreg = knn[(size_t)p * 32 + kbase + (lane & 15)];

    const float* cx = vc + (size_t)b * 3 * N_;
    const int j0 = __shfl(jreg, 0, 32);
    const float scx = cx[j0], scy = cx[N_ + j0], scz = cx[2 * N_ + j0];
    const float cnt = (float)((size_t)B_ * N_ * 16);

    // per-o BN params live in lane o (o = lane&15)
    const int o = lane & 15;
    const float tm   = tstats[o] / cnt;
    const float tvar = tstats[16 + o] / cnt - tm * tm;
    const float tinv = rsqrtf(tvar + EPSV) * cg[o];
    const float tsh  = cb[o] - tm * tinv;

    // w[k]: lane computes k = lane&15 over 8 of the 16 o's
    const float cnx = cx[jreg], cny = cx[N_ + jreg], cnz = cx[2 * N_ + jreg];
    float wacc = 0.f;
    const int obase = (lane >> 4) * 8;
#pragma unroll
    for (int oo = 0; oo < 8; ++oo) {
        const int oc = obase + oo;
        const float* w = cw + oc * 6;
        const float t = w[0] * scx + w[1] * scy + w[2] * scz
                      + w[3] * (scx - cnx) + w[4] * (scy - cny) + w[5] * (scz - cnz);
        wacc += tanhf(t * __shfl(tinv, oc, 32) + __shfl(tsh, oc, 32));
    }
    wacc += __shfl_xor(wacc, 16, 32);    // lanes k and k+16 both hold w[k]

    // per-channel BN params: e = lane and e = lane+32
    const float hm0 = hstats[lane] / cnt;
    const float hi0 = rsqrtf(hstats[DOUT + lane] / cnt - hm0 * hm0 + EPSV) * fg[lane];
    const float hs0 = fb[lane] - hm0 * hi0;
    const float hm1 = hstats[lane + 32] / cnt;
    const float hi1 = rsqrtf(hstats[DOUT + 32 + lane] / cnt - hm1 * hm1 + EPSV) * fg[lane + 32];
    const float hs1 = fb[lane + 32] - hm1 * hi1;

    const float* Zb = Z + (size_t)b * N_ * DOUT;
    float acc0 = 0.f, acc1 = 0.f;
#pragma unroll
    for (int kk = 0; kk < 16; ++kk) {
        const float wk = __shfl(wacc, kk, 32);
        const int j = __shfl(jreg, kk, 32);
        const float* zr = Zb + (size_t)j * DOUT;
        acc0 += wk * fmaxf(zr[lane] * hi0 + hs0, 0.f);
        acc1 += wk * fmaxf(zr[lane + 32] * hi1 + hs1, 0.f);
    }
    float* ob = out + (size_t)b * DOUT * N_ + n;
    ob[(size_t)lane * N_] += acc0;
    ob[(size_t)(lane + 32) * N_] += acc1;
}

// ---------------------------------------------------------------------------
extern "C" void kernel_launch(void* const* d_in, const int* in_sizes, int n_in,
                              void* d_out, int out_size, void* d_ws, size_t ws_size,
                              hipStream_t stream)
{
    const float* vf    = (const float*)d_in[0];
    const float* vc    = (const float*)d_in[1];
    const float* fs_w  = (const float*)d_in[2];
    const float* fs_b  = (const float*)d_in[3];
    const float* fs_g  = (const float*)d_in[4];
    const float* fs_be = (const float*)d_in[5];
    const float* fr_w0 = (const float*)d_in[6];
    const float* fr_g0 = (const float*)d_in[7];
    const float* fr_be0= (const float*)d_in[8];
    const float* cr_w0 = (const float*)d_in[9];
    const float* cr_g0 = (const float*)d_in[10];
    const float* cr_be0= (const float*)d_in[11];
    const float* fr_w1 = (const float*)d_in[12];
    const float* fr_g1 = (const float*)d_in[13];
    const float* fr_be1= (const float*)d_in[14];
    const float* cr_w1 = (const float*)d_in[15];
    const float* cr_g1 = (const float*)d_in[16];
    const float* cr_be1= (const float*)d_in[17];

    char* ws = (char*)d_ws;
    float* Y    = (float*)(ws);                              // 8 MB
    float* Z0   = (float*)(ws + (size_t)8  * 1024 * 1024);   // 8 MB
    float* Z1   = (float*)(ws + (size_t)16 * 1024 * 1024);   // 8 MB
    int*   knn  = (int*)  (ws + (size_t)24 * 1024 * 1024);   // 4 MB
    float* st   = (float*)(ws + (size_t)28 * 1024 * 1024);   // 448 floats
    float* fsSt = st;
    float* h0   = st + 128;
    float* t0   = st + 256;
    float* h1   = st + 288;
    float* t1   = st + 416;
    float* out  = (float*)d_out;

    hipMemsetAsync(st, 0, 448 * sizeof(float), stream);

    k_gemm3<<<3072, 256, 0, stream>>>(vf, fs_w, fr_w0, fr_w1, fs_b, Y, Z0, Z1);
    k_fs_stats<<<DOUT, 256, 0, stream>>>(Y, fsSt);
    k_base<<<(B_ * DOUT * N_) / 256, 256, 0, stream>>>(Y, fsSt, fs_g, fs_be, out);
    k_knn<<<B_ * (N_ / QT), QT, 0, stream>>>(vc, knn);

    k_layer_stats<<<(B_ * N_) / 8, 256, 0, stream>>>(Z0, knn, vc, cr_w0, 0, h0, t0);
    k_layer_apply<<<(B_ * N_) / 8, 256, 0, stream>>>(Z0, knn, vc, cr_w0, cr_g0, cr_be0,
                                                     fr_g0, fr_be0, h0, t0, 0, out);
    k_layer_stats<<<(B_ * N_) / 8, 256, 0, stream>>>(Z1, knn, vc, cr_w1, 16, h1, t1);
    k_layer_apply<<<(B_ * N_) / 8, 256, 0, stream>>>(Z1, knn, vc, cr_w1, cr_g1, cr_be1,
                                                     fr_g1, fr_be1, h1, t1, 16, out);
}